// LSTM_11003706213031
// MI455X (gfx1250) — compile-verified
//
#include <hip/hip_runtime.h>
#include <hip/hip_bf16.h>

// CDNA5 / gfx1250 projected-LSTM, fp32 WMMA (V_WMMA_F32_16X16X4_F32).
// One workgroup (4 waves) = 16 batch rows; wave w owns gate tiles {w,w+4,w+8,w+12}
// so i/f/g/o for the same hidden columns stay in-wave and c is register-resident.
// All weight B-fragments are register-resident (loaded once from global);
// LDS carries only the per-step h/a cross-wave exchange (8 KB).

typedef __attribute__((ext_vector_type(2))) float v2f;
typedef __attribute__((ext_vector_type(8))) float v8f;

#define T_STEPS 512
#define IN      4
#define HID     64
#define PROJ    52
#define G       256   // 4*HID gate width
#define KW      56    // IN + PROJ  (Z = [x_t | h])
#define KC      14    // KW / 4 K-chunks
#define ROWS    16    // batch rows per workgroup
#define NBATCH  4096

#if __has_builtin(__builtin_amdgcn_rcpf)
#define FRCP(x) __builtin_amdgcn_rcpf(x)
#else
#define FRCP(x) (1.0f / (x))
#endif
#if __has_builtin(__builtin_amdgcn_exp2f)
#define FEXP2(x) __builtin_amdgcn_exp2f(x)
#else
#define FEXP2(x) exp2f(x)
#endif

#define LOG2E 1.4426950408889634f

static __device__ __forceinline__ v8f wmma_f32_16x16x4(v2f a, v2f b, v8f c) {
  // (neg_a, A, neg_b, B, c_mod, C, reuse_a, reuse_b)
  return __builtin_amdgcn_wmma_f32_16x16x4_f32(false, a, false, b, (short)0, c,
                                               false, false);
}

static __device__ __forceinline__ v8f splat8(float s) {
  v8f v;
#pragma unroll
  for (int i = 0; i < 8; ++i) v[i] = s;
  return v;
}

// sigmoid via v_exp_f32 + v_rcp_f32 (exp2(-x*log2e) never overflows usefully)
static __device__ __forceinline__ float fast_sig(float x) {
  return FRCP(1.0f + FEXP2(x * -LOG2E));
}

// tanh via v_tanh_f32 if available, else overflow-safe exp2/rcp form
static __device__ __forceinline__ float fast_tanh(float x) {
#if __has_builtin(__builtin_amdgcn_tanhf)
  return __builtin_amdgcn_tanhf(x);
#else
  const float ax = __builtin_fabsf(x);
  const float e  = FEXP2(ax * (-2.0f * LOG2E));   // exp(-2|x|) in (0,1]
  const float r  = (1.0f - e) * FRCP(1.0f + e);
  return __builtin_copysignf(r, x);
#endif
}

__global__ void __launch_bounds__(128, 1)
lstm_proj_wmma_kernel(const float* __restrict__ x,     // [4096][512][4]
                      const float* __restrict__ W_ih,  // [256][4]
                      const float* __restrict__ W_hh,  // [256][52]
                      const float* __restrict__ b_ih,  // [256]
                      const float* __restrict__ b_hh,  // [256]
                      const float* __restrict__ W_hr,  // [52][64]
                      float* __restrict__ out)         // [4096][52]
{
  __shared__ float abuf[ROWS * HID];   // [16][64] o*tanh(c)
  __shared__ float hbuf[ROWS * HID];   // [16][64] projected h (cols 52..63 = 0)

  const int tid   = threadIdx.x;
  const int wave  = tid >> 5;          // 0..3
  const int lane  = tid & 31;
  const int l16   = lane & 15;
  const int hi    = lane >> 4;         // 0 | 1
  const int dlt   = hi * 2;            // K offset within a 4-wide chunk
  const int rbase = blockIdx.x * ROWS;

  // h0 = 0 (and clear abuf)
  for (int idx = tid; idx < ROWS * HID; idx += blockDim.x) {
    abuf[idx] = 0.0f;
    hbuf[idx] = 0.0f;
  }

  // ---- per-wave register-resident constants ----
  float gbias[4];
  int   coln[4];
#pragma unroll
  for (int j = 0; j < 4; ++j) {               // j: 0=i 1=f 2=g 3=o
    const int col = 16 * (wave + 4 * j) + l16;
    coln[j]  = col;
    gbias[j] = b_ih[col] + b_hh[col];
  }

  // Gate-weight B-fragments, register resident (112 VGPRs).
  // Z row r, gate col c:  r<4 -> W_ih[c][r],  r>=4 -> W_hh[c][r-4]
  v2f wtB[KC][4];
#pragma unroll
  for (int j = 0; j < 4; ++j) {
    const int col = coln[j];
    wtB[0][j].x = W_ih[col * IN + dlt];
    wtB[0][j].y = W_ih[col * IN + dlt + 1];
#pragma unroll
    for (int k = 1; k < KC; ++k) {
      const int hr = 4 * k + dlt - IN;        // 0..51
      wtB[k][j].x = W_hh[col * PROJ + hr];
      wtB[k][j].y = W_hh[col * PROJ + hr + 1];
    }
  }

  // W_hr^T B-fragments for projection tile n = wave (zero-padded cols >= 52)
  v2f whrB[16];
  {
    const int pcol = 16 * wave + l16;
#pragma unroll
    for (int k = 0; k < 16; ++k) {
      const int kk = 4 * k + dlt;
      v2f b;
      b.x = (pcol < PROJ) ? W_hr[pcol * HID + kk]     : 0.0f;
      b.y = (pcol < PROJ) ? W_hr[pcol * HID + kk + 1] : 0.0f;
      whrB[k] = b;
    }
  }

  const float* xrow =
      x + (size_t)(rbase + l16) * T_STEPS * IN + dlt;   // A-frag source, chunk 0

  v8f c = splat8(0.0f);                                 // cell state tile
  __syncthreads();

  for (int t = 0; t < T_STEPS; ++t) {
    // ================= gates: Z[16,56] x W[56,256] + b =================
    v8f acc[4];
#pragma unroll
    for (int j = 0; j < 4; ++j) acc[j] = splat8(gbias[j]);

#pragma unroll
    for (int k = 0; k < KC; ++k) {
      v2f a;
      if (k == 0) {                                    // x_t from global
        const float* xp = xrow + t * IN;
        a.x = xp[0]; a.y = xp[1];
      } else {                                         // h from LDS
        a = *(const v2f*)&hbuf[l16 * HID + (4 * k - IN) + dlt];
      }
#pragma unroll
      for (int j = 0; j < 4; ++j)
        acc[j] = wmma_f32_16x16x4(a, wtB[k][j], acc[j]);
    }
    __builtin_prefetch(xrow + (t + 1) * IN, 0, 1);     // global_prefetch_b8

    // ================= elementwise cell update (in-wave) ================
#pragma unroll
    for (int r = 0; r < 8; ++r) {
      const float iv = fast_sig(acc[0][r]);
      const float fv = fast_sig(acc[1][r]);
      const float gv = fast_tanh(acc[2][r]);
      const float ov = fast_sig(acc[3][r]);
      const float cv = fv * c[r] + iv * gv;
      c[r] = cv;
      abuf[(r + 8 * hi) * HID + 16 * wave + l16] = ov * fast_tanh(cv);
    }
    __syncthreads();

    // ============ projection: h = a[16,64] x W_hr^T[64,52] =============
    v8f p = splat8(0.0f);
#pragma unroll
    for (int k = 0; k < 16; ++k) {
      v2f pa = *(const v2f*)&abuf[l16 * HID + 4 * k + dlt];
      p = wmma_f32_16x16x4(pa, whrB[k], p);
    }
#pragma unroll
    for (int r = 0; r < 8; ++r)
      hbuf[(r + 8 * hi) * HID + 16 * wave + l16] = p[r];
    __syncthreads();
  }

  // ---- emit h_last [16][52] ----
  for (int idx = tid; idx < ROWS * PROJ; idx += blockDim.x) {
    const int r = idx / PROJ, cc = idx - r * PROJ;
    out[(size_t)(rbase + r) * PROJ + cc] = hbuf[r * HID + cc];
  }
}

extern "C" void kernel_launch(void* const* d_in, const int* in_sizes, int n_in,
                              void* d_out, int out_size, void* d_ws, size_t ws_size,
                              hipStream_t stream) {
  (void)in_sizes; (void)n_in; (void)out_size; (void)d_ws; (void)ws_size;
  const float* x    = (const float*)d_in[0];
  const float* W_ih = (const float*)d_in[1];
  const float* W_hh = (const float*)d_in[2];
  const float* b_ih = (const float*)d_in[3];
  const float* b_hh = (const float*)d_in[4];
  const float* W_hr = (const float*)d_in[5];
  float* out = (float*)d_out;

  dim3 grid(NBATCH / ROWS);   // 256 workgroups (~1 per WGP)
  dim3 block(128);            // 4 waves (wave32)
  lstm_proj_wmma_kernel<<<grid, block, 0, stream>>>(x, W_ih, W_hh, b_ih, b_hh,
                                                    W_hr, out);
}